// NSF_AR_34866544509308
// MI455X (gfx1250) — compile-verified
//
#include <hip/hip_runtime.h>
#include <math.h>

// ---------------- problem constants ----------------
#define KBINS   5
#define DIMS    64
#define NPAR    14          // 3K-1
#define MINBW   1e-3f
#define MIND    1e-3f

typedef float v2f __attribute__((ext_vector_type(2)));
typedef float v8f __attribute__((ext_vector_type(8)));

#define WPB      4          // waves per block
#define XSTRIDE  66         // 32-row x tile, padded (bank-conflict-free b64)
#define HSTRIDE  18         // 32-row scratch tile (h / spline params)

#define WB() __builtin_amdgcn_wave_barrier()

static __device__ __forceinline__ v8f wmma4(v2f a, v2f b, v8f c) {
    // D = A(16x4 f32) * B(4x16 f32) + C(16x16 f32)
    return __builtin_amdgcn_wmma_f32_16x16x4_f32(
        false, a, false, b, (short)0, c, false, false);
}

static __device__ __forceinline__ float sp_(float x) {
    return (x > 20.f) ? x : log1pf(expf(x));
}

static __device__ __forceinline__ void softmax5(const float* a, float* o) {
    float m = a[0];
#pragma unroll
    for (int k = 1; k < 5; ++k) m = fmaxf(m, a[k]);
    float s = 0.f;
#pragma unroll
    for (int k = 0; k < 5; ++k) { o[k] = expf(a[k] - m); s += o[k]; }
    float r = 1.f / s;
#pragma unroll
    for (int k = 0; k < 5; ++k) o[k] *= r;
}

// rational-quadratic spline, forward; mirrors the reference exactly
static __device__ __forceinline__ void rqs(const float* p, float x,
                                           float& z, float& ld) {
    const float Bc = 3.0f;
    bool inside = (x >= -Bc) && (x <= Bc);
    float xc = fminf(fmaxf(x, -Bc), Bc);

    float s0[5], s1[5], W[5], H[5], ws[5], hs[5];
    softmax5(p, s0);                 // softmax of W logits
    softmax5(p + 5, s1);             // softmax of H logits
#pragma unroll
    for (int k = 0; k < 5; ++k) { W[k] = 6.f * s0[k]; H[k] = 6.f * s1[k]; }
    softmax5(W, ws);                 // re-softmax inside RQS (as reference)
    softmax5(H, hs);

    float cw[6], ch[6];
    cw[0] = -Bc; ch[0] = -Bc;
    float aw = 0.f, ah = 0.f;
#pragma unroll
    for (int k = 0; k < 5; ++k) {
        aw += MINBW + (1.f - 5.f * MINBW) * ws[k];
        ah += MINBW + (1.f - 5.f * MINBW) * hs[k];
        cw[k + 1] = 6.f * aw - 3.f;
        ch[k + 1] = 6.f * ah - 3.f;
    }
    cw[5] = Bc; ch[5] = Bc;

    float dv[6];
    dv[0] = 1.0f; dv[5] = 1.0f;      // MIN_D + softplus(DERIV_CONST) == 1
#pragma unroll
    for (int k = 1; k < 5; ++k) dv[k] = MIND + sp_(sp_(p[9 + k]));

    int b = 0;
#pragma unroll
    for (int k = 0; k < 5; ++k) b += (xc >= cw[k]) ? 1 : 0;
    b += (xc >= Bc + 1e-6f) ? 1 : 0;
    b -= 1; b = b < 0 ? 0 : (b > 4 ? 4 : b);

    float icw = cw[0], iw = cw[1] - cw[0];
    float ich = ch[0], ih = ch[1] - ch[0];
    float dk = dv[0], dk1 = dv[1];
#pragma unroll
    for (int k = 1; k < 5; ++k) {
        if (b == k) {
            icw = cw[k]; iw = cw[k + 1] - cw[k];
            ich = ch[k]; ih = ch[k + 1] - ch[k];
            dk = dv[k]; dk1 = dv[k + 1];
        }
    }
    float delta = ih / iw;
    float th = (xc - icw) / iw;
    float tt = th * (1.f - th);
    float num = ih * (delta * th * th + dk * tt);
    float den = delta + (dk + dk1 - 2.f * delta) * tt;
    float outv = ich + num / den;
    float dnum = delta * delta *
                 (dk1 * th * th + 2.f * delta * tt + dk * (1.f - th) * (1.f - th));
    float l = logf(dnum) - 2.f * logf(den);
    z  = inside ? outv : x;
    ld = inside ? l : 0.f;
}

// ---------------- weight pre-pack (B-fragment layout) ----------------
struct PtrArgs { const float* p[378]; };

__global__ void nsf_pack(PtrArgs args, float* __restrict__ ws) {
    int j = blockIdx.x;          // layer 0..62, conditioner input width i=j+1
    int i = j + 1;
    int T1 = (i + 3) >> 2;
    long base = 0;
    for (int l = 0; l < j; ++l) base += (long)(((l + 4) >> 2) * 64 + 304);
    const float* w1 = args.p[6 * j + 0];
    const float* b1 = args.p[6 * j + 1];
    const float* w2 = args.p[6 * j + 2];
    const float* b2 = args.p[6 * j + 3];
    const float* w3 = args.p[6 * j + 4];
    const float* b3 = args.p[6 * j + 5];
    int count = T1 * 64 + 304;
    for (int idx = threadIdx.x; idx < count; idx += blockDim.x) {
        int r = idx; float val = 0.f;
        if (r < T1 * 64) {                               // W1 fragments
            int t = r >> 6, q = r & 63, l = q >> 1, e = q & 1;
            int n = l & 15, k = t * 4 + (l >> 4) * 2 + e;
            val = (k < i && n < 8) ? w1[k * 8 + n] : 0.f;
        } else { r -= T1 * 64;
            if (r < 16) { val = (r < 8) ? b1[r] : 0.f; } // b1 (16-padded)
            else { r -= 16;
                if (r < 128) {                            // W2 fragments (K=8)
                    int t = r >> 6, q = r & 63, l = q >> 1, e = q & 1;
                    int n = l & 15, k = t * 4 + (l >> 4) * 2 + e;
                    val = (n < 8) ? w2[k * 8 + n] : 0.f;
                } else { r -= 128;
                    if (r < 16) { val = (r < 8) ? b2[r] : 0.f; }
                    else { r -= 16;
                        if (r < 128) {                    // W3 fragments (K=8,N=14)
                            int t = r >> 6, q = r & 63, l = q >> 1, e = q & 1;
                            int n = l & 15, k = t * 4 + (l >> 4) * 2 + e;
                            val = (n < 14) ? w3[k * 14 + n] : 0.f;
                        } else { r -= 128;
                            val = (r < 14) ? b3[r] : 0.f; // b3 (16-padded)
                        }
                    }
                }
            }
        }
        ws[base + idx] = val;
    }
}

// ---------------- main fused kernel ----------------
__launch_bounds__(32 * WPB)
__global__ void nsf_main(const float* __restrict__ x,
                         const float* __restrict__ initp,
                         const float* __restrict__ wpk,
                         float* __restrict__ zout,
                         float* __restrict__ ldout) {
    __shared__ float xs[WPB][32 * XSTRIDE];
    __shared__ float hb[WPB][32 * HSTRIDE];
    const int lane = threadIdx.x & 31;
    const int wid  = threadIdx.x >> 5;
    const int rowbase = (blockIdx.x * WPB + wid) * 32;
    float* xw = xs[wid];
    float* hw = hb[wid];

    // stage 32x64 x-tile into LDS (coalesced b64 loads)
    const v2f* x2 = (const v2f*)(x + (size_t)rowbase * DIMS);
#pragma unroll
    for (int u = 0; u < 32; ++u) {
        int linear = u * 32 + lane;
        int r = linear >> 5, c2 = linear & 31;
        v2f v = x2[r * 32 + c2];
        *(v2f*)&xw[r * XSTRIDE + c2 * 2] = v;
    }
    WB();

    const int n = lane & 15;
    const int half = lane >> 4;
    float ldsum = 0.f;

    // dim 0: broadcast init params
    {
        float p[NPAR];
#pragma unroll
        for (int c = 0; c < NPAR; ++c) p[c] = initp[c];
        float xv = xw[lane * XSTRIDE + 0];
        float z, ld; rqs(p, xv, z, ld);
        zout[(size_t)(rowbase + lane) * DIMS + 0] = z;
        ldsum += ld;
    }

    const float* wp = wpk;
    const v8f zero = {0, 0, 0, 0, 0, 0, 0, 0};
    for (int j = 0; j < 63; ++j) {
        const int i = j + 1;
        const int T1 = (i + 3) >> 2;
        // ---- layer 1: x[:, :i] @ W1 ----
        v8f acc0 = zero, acc1 = zero;
        const v2f* bfr = (const v2f*)wp;
        for (int t = 0; t < T1; ++t) {
            v2f b  = bfr[t * 32 + lane];
            v2f a0 = *(const v2f*)&xw[n * XSTRIDE + t * 4 + half * 2];
            v2f a1 = *(const v2f*)&xw[(16 + n) * XSTRIDE + t * 4 + half * 2];
            acc0 = wmma4(a0, b, acc0);
            acc1 = wmma4(a1, b, acc1);
        }
        wp += T1 * 64;
        float bias = wp[n]; wp += 16;
        WB();
#pragma unroll
        for (int v = 0; v < 8; ++v) {
            int m = half * 8 + v;
            hw[m * HSTRIDE + n]        = tanhf(acc0[v] + bias);
            hw[(16 + m) * HSTRIDE + n] = tanhf(acc1[v] + bias);
        }
        WB();
        // ---- layer 2: h @ W2 (K=8) ----
        acc0 = zero; acc1 = zero;
        bfr = (const v2f*)wp;
#pragma unroll
        for (int t = 0; t < 2; ++t) {
            v2f b  = bfr[t * 32 + lane];
            v2f a0 = *(const v2f*)&hw[n * HSTRIDE + t * 4 + half * 2];
            v2f a1 = *(const v2f*)&hw[(16 + n) * HSTRIDE + t * 4 + half * 2];
            acc0 = wmma4(a0, b, acc0);
            acc1 = wmma4(a1, b, acc1);
        }
        wp += 128;
        bias = wp[n]; wp += 16;
        WB();
#pragma unroll
        for (int v = 0; v < 8; ++v) {
            int m = half * 8 + v;
            hw[m * HSTRIDE + n]        = tanhf(acc0[v] + bias);
            hw[(16 + m) * HSTRIDE + n] = tanhf(acc1[v] + bias);
        }
        WB();
        // ---- layer 3: h @ W3 (K=8, 14 outputs) ----
        acc0 = zero; acc1 = zero;
        bfr = (const v2f*)wp;
#pragma unroll
        for (int t = 0; t < 2; ++t) {
            v2f b  = bfr[t * 32 + lane];
            v2f a0 = *(const v2f*)&hw[n * HSTRIDE + t * 4 + half * 2];
            v2f a1 = *(const v2f*)&hw[(16 + n) * HSTRIDE + t * 4 + half * 2];
            acc0 = wmma4(a0, b, acc0);
            acc1 = wmma4(a1, b, acc1);
        }
        wp += 128;
        bias = wp[n]; wp += 16;
        WB();
#pragma unroll
        for (int v = 0; v < 8; ++v) {
            int m = half * 8 + v;
            hw[m * HSTRIDE + n]        = acc0[v] + bias;
            hw[(16 + m) * HSTRIDE + n] = acc1[v] + bias;
        }
        WB();
        // ---- spline: one row per lane ----
        float p[NPAR];
#pragma unroll
        for (int c = 0; c < NPAR; ++c) p[c] = hw[lane * HSTRIDE + c];
        float xv = xw[lane * XSTRIDE + i];
        float z, ld; rqs(p, xv, z, ld);
        zout[(size_t)(rowbase + lane) * DIMS + i] = z;
        ldsum += ld;
        WB();
    }
    ldout[rowbase + lane] = ldsum;
}

// ---------------- launcher ----------------
extern "C" void kernel_launch(void* const* d_in, const int* in_sizes, int n_in,
                              void* d_out, int out_size, void* d_ws, size_t ws_size,
                              hipStream_t stream) {
    const float* x     = (const float*)d_in[0];
    const float* initp = (const float*)d_in[1];
    if (n_in < 2 + 63 * 6) return;

    PtrArgs pa;
    for (int j = 0; j < 378; ++j) pa.p[j] = (const float*)d_in[2 + j];

    float* ws  = (float*)d_ws;
    float* z   = (float*)d_out;
    int nrows  = in_sizes[0] / DIMS;
    float* ldo = z + (size_t)nrows * DIMS;

    nsf_pack<<<dim3(63), dim3(128), 0, stream>>>(pa, ws);
    nsf_main<<<dim3(nrows / (32 * WPB)), dim3(32 * WPB), 0, stream>>>(
        x, initp, ws, z, ldo);
}